// DiffeqSolver_24275155157251
// MI455X (gfx1250) — compile-verified
//
#include <hip/hip_runtime.h>
#include <cstddef>

// Problem constants (match reference)
#define BB  16
#define NN  512
#define DD  128
#define TT  20
#define BND (BB * NN * DD)   // 1,048,576 floats per state

typedef __attribute__((ext_vector_type(2))) float v2f;
typedef __attribute__((ext_vector_type(8))) float v8f;

// ---- gfx1250 async global->LDS staging (ASYNCcnt path), with safe fallback ----
#if defined(__has_builtin)
#  if __has_builtin(__builtin_amdgcn_global_load_async_to_lds_b128)
#    define HAVE_ASYNC_LDS 1
#  endif
#endif
#ifndef HAVE_ASYNC_LDS
#  define HAVE_ASYNC_LDS 0
#endif

__device__ __forceinline__ void stage16(const float* __restrict__ g, float* l) {
#if HAVE_ASYNC_LDS
    // Param types per hipcc diagnostic: AS(1) pointer to vector_size(16) int.
    typedef int v4i_ __attribute__((vector_size(16)));
    typedef __attribute__((address_space(1))) v4i_ gv4;
    typedef __attribute__((address_space(3))) v4i_ lv4;
    // per-lane 16B copy from global address g to LDS address l, tracked by ASYNCcnt
    __builtin_amdgcn_global_load_async_to_lds_b128((gv4*)(void*)g, (lv4*)(void*)l,
                                                   /*imm offset=*/0, /*cpol=*/0);
#else
    *(float4*)l = *(const float4*)g;
#endif
}

__device__ __forceinline__ void async_publish_wait() {
#if HAVE_ASYNC_LDS
#  if __has_builtin(__builtin_amdgcn_s_wait_asynccnt)
    __builtin_amdgcn_s_wait_asynccnt(0);
#  else
    asm volatile("s_wait_asynccnt 0x0" ::: "memory");
#  endif
#endif
}

// CDNA5 f32 WMMA: D(16x16,f32) = A(16x4,f32) * B(4x16,f32) + C
__device__ __forceinline__ v8f wmma_f32(v2f a, v2f b, v8f c) {
    return __builtin_amdgcn_wmma_f32_16x16x4_f32(
        /*neg_a=*/false, a, /*neg_b=*/false, b,
        /*c_mod=*/(short)0, c, /*reuse_a=*/false, /*reuse_b=*/false);
}

// d_out[0] = node  (x at t0)
__global__ void __launch_bounds__(256)
init_copy_kernel(const float* __restrict__ src, float* __restrict__ dst) {
    int i = (blockIdx.x * 256 + threadIdx.x) * 4;   // BND/4 float4s
    *(float4*)(dst + i) = *(const float4*)(src + i);
}

// One RK4 stage:  k = tanh((E·x_in)W1 + x_in·W2 + b)  fused with the
// stage-specific elementwise update against x_base / acc.
//   STAGE 0: acc  = k;       x_out = x_base + 0.5*dt*k
//   STAGE 1: acc += 2k;      x_out = x_base + 0.5*dt*k
//   STAGE 2: acc += 2k;      x_out = x_base +     dt*k
//   STAGE 3:                 x_out = x_base + dt/6*(acc + k)
template <int STAGE>
__global__ void __launch_bounds__(256)
rk4_stage_kernel(const float* __restrict__ edge,    // [B,N,N]
                 const float* __restrict__ x_in,    // [B,N,D] stage input
                 const float* __restrict__ x_base,  // [B,N,D] step base x_t
                 const float* __restrict__ W1,      // [D,D]
                 const float* __restrict__ W2,      // [D,D]
                 const float* __restrict__ bias,    // [D]
                 const float* __restrict__ ts,      // [T]
                 int step,
                 float* __restrict__ x_out,         // [B,N,D]
                 float* __restrict__ acc)           // [B,N,D] k-accumulator
{
    // Padded LDS (strides 516/136 floats -> 64-bank conflict-free; byte strides
    // 2064/544 are 16B multiples so async b128 lane destinations stay aligned)
    __shared__ __align__(16) float sE [16][516];  // E tile (phase 1); agg (phase 2)
    __shared__ __align__(16) float sXc[32][136];  // X K-chunk (phase 1)
    __shared__ __align__(16) float sXr[16][136];  // x_in row tile (phase 2)

    const int tid  = threadIdx.x;
    const int wave = tid >> 5;          // 8 wave32s, each owns 16 cols of D
    const int lane = tid & 31;
    const int b    = blockIdx.x >> 5;   // 32 row-tiles per batch
    const int row0 = (blockIdx.x & 31) << 4;
    const int c0   = wave << 4;

    const size_t boff  = (size_t)b * NN * DD;
    const float* xin_b = x_in + boff;

    // ---- async stage: E tile 16x512 ----
    {
        const float* esrc = edge + ((size_t)b * NN + row0) * NN;
#pragma unroll
        for (int i = 0; i < 8; ++i) {
            int f = tid + i * 256;            // float4 index in [0,2048)
            int r = f >> 7;
            int c = (f & 127) << 2;
            stage16(esrc + (size_t)r * NN + c, &sE[r][c]);
        }
    }
    // ---- async stage: x_in row tile 16x128 ----
    {
#pragma unroll
        for (int i = 0; i < 2; ++i) {
            int f = tid + i * 256;            // [0,512)
            int r = f >> 5;
            int c = (f & 31) << 2;
            stage16(xin_b + (size_t)(row0 + r) * DD + c, &sXr[r][c]);
        }
    }

    // WMMA f32 lane mapping (ISA 7.12.2):
    //   A 16x4 : lane m (0-15) holds A[m][0..1]; lane 16+m holds A[m][2..3]
    //   B 4x16 : lane n (0-15) holds B[0..1][n]; lane 16+n holds B[2..3][n]
    //   C 16x16: vgpr r, lanes 0-15 -> (M=r, N=lane); lanes 16-31 -> (M=r+8)
    const int am = lane & 15;
    const int ak = (lane >> 4) << 1;          // 0 or 2

    v8f aggf = {0.f, 0.f, 0.f, 0.f, 0.f, 0.f, 0.f, 0.f};

    // ---- Phase 1: agg = E_rows(16x512) @ X_b(512x128), K chunked by 32 ----
    for (int ch = 0; ch < 16; ++ch) {
        __syncthreads();                      // prior chunk's LDS reads retired
#pragma unroll
        for (int i = 0; i < 4; ++i) {
            int f = tid + i * 256;            // [0,1024)
            int r = f >> 5;
            int c = (f & 31) << 2;
            stage16(xin_b + (size_t)(ch * 32 + r) * DD + c, &sXc[r][c]);
        }
        async_publish_wait();                 // ASYNCcnt==0 (covers E/Xr on ch==0)
        __syncthreads();
#pragma unroll
        for (int kk = 0; kk < 32; kk += 4) {
            v2f a, bf;
            a.x  = sE [am][ch * 32 + kk + ak];
            a.y  = sE [am][ch * 32 + kk + ak + 1];
            bf.x = sXc[kk + ak    ][c0 + am];
            bf.y = sXc[kk + ak + 1][c0 + am];
            aggf = wmma_f32(a, bf, aggf);
        }
    }

    __syncthreads();
    // spill agg fragment into sE[0..15][0..127] (tile shared by all 8 waves)
    {
        const int cn = lane & 15;
        const int rb = (lane >> 4) << 3;
#pragma unroll
        for (int r = 0; r < 8; ++r)
            sE[rb + r][c0 + cn] = aggf[r];
    }
    __syncthreads();

    // ---- Phase 2: out = agg@W1 + x_rows@W2 (K = 128), W1/W2 from L2 ----
    v8f of = {0.f, 0.f, 0.f, 0.f, 0.f, 0.f, 0.f, 0.f};
#pragma unroll 4
    for (int k = 0; k < DD; k += 4) {
        v2f a1, a2, b1, b2;
        a1.x = sE [am][k + ak];  a1.y = sE [am][k + ak + 1];
        a2.x = sXr[am][k + ak];  a2.y = sXr[am][k + ak + 1];
        b1.x = W1[(size_t)(k + ak) * DD + c0 + am];
        b1.y = W1[(size_t)(k + ak + 1) * DD + c0 + am];
        b2.x = W2[(size_t)(k + ak) * DD + c0 + am];
        b2.y = W2[(size_t)(k + ak + 1) * DD + c0 + am];
        of = wmma_f32(a1, b1, of);
        of = wmma_f32(a2, b2, of);
    }

    // ---- Epilogue: tanh + RK4 stage update ----
    const float dt = ts[step + 1] - ts[step];
    const int cn = lane & 15;
    const int rb = (lane >> 4) << 3;
    const float bv = bias[c0 + cn];
#pragma unroll
    for (int r = 0; r < 8; ++r) {
        const int row = row0 + rb + r;
        const size_t idx = boff + (size_t)row * DD + c0 + cn;
        const float kv  = tanhf(of[r] + bv);
        const float xb0 = x_base[idx];
        if (STAGE == 0) {
            acc[idx]   = kv;
            x_out[idx] = xb0 + 0.5f * dt * kv;
        } else if (STAGE == 1) {
            acc[idx]  += 2.0f * kv;
            x_out[idx] = xb0 + 0.5f * dt * kv;
        } else if (STAGE == 2) {
            acc[idx]  += 2.0f * kv;
            x_out[idx] = xb0 + dt * kv;
        } else {
            x_out[idx] = xb0 + (dt * (1.0f / 6.0f)) * (acc[idx] + kv);
        }
    }
}

extern "C" void kernel_launch(void* const* d_in, const int* in_sizes, int n_in,
                              void* d_out, int out_size, void* d_ws, size_t ws_size,
                              hipStream_t stream) {
    (void)in_sizes; (void)n_in; (void)out_size; (void)ws_size;
    const float* node = (const float*)d_in[0];
    const float* edge = (const float*)d_in[1];
    const float* ts   = (const float*)d_in[2];
    const float* W1   = (const float*)d_in[3];
    const float* W2   = (const float*)d_in[4];
    const float* bias = (const float*)d_in[5];
    float* out = (float*)d_out;                 // [T,B,N,D]

    float* acc = (float*)d_ws;                  // 4 MB k-accumulator
    float* xA  = acc + BND;                     // stage ping
    float* xB  = xA + BND;                      // stage pong

    // pred_y[0] = node
    init_copy_kernel<<<BND / 1024, 256, 0, stream>>>(node, out);

    dim3 grid(BB * (NN / 16));                  // 512 blocks, 256 thr (8 wave32)
    for (int t = 0; t < TT - 1; ++t) {
        const float* xb = out + (size_t)t * BND;       // x_t lives in d_out[t]
        float* xnext    = out + (size_t)(t + 1) * BND; // x_{t+1} -> d_out[t+1]
        rk4_stage_kernel<0><<<grid, 256, 0, stream>>>(edge, xb, xb, W1, W2, bias, ts, t, xA, acc);
        rk4_stage_kernel<1><<<grid, 256, 0, stream>>>(edge, xA, xb, W1, W2, bias, ts, t, xB, acc);
        rk4_stage_kernel<2><<<grid, 256, 0, stream>>>(edge, xB, xb, W1, W2, bias, ts, t, xA, acc);
        rk4_stage_kernel<3><<<grid, 256, 0, stream>>>(edge, xA, xb, W1, W2, bias, ts, t, xnext, acc);
    }
}